// AutoregressiveBisectionInverter_28432683499923
// MI455X (gfx1250) — compile-verified
//
#include <hip/hip_runtime.h>
#include <hip/hip_bf16.h>
#include <math.h>

// CDNA5 / gfx1250, wave32. One wave per 32 batch rows.
// Panel updates via V_WMMA_F32_16X16X4_F32 (f32 precision required: bisection
// converges to +-1e-6, bf16 operands would inject ~1e-3 error into c_i).

typedef __attribute__((ext_vector_type(2))) float v2f;
typedef __attribute__((ext_vector_type(8))) float v8f;

#define D64   64
#define ROWS  32      // batch rows per workgroup (one wave)
#define WPAD  65      // LDS row pitch (floats) to dodge bank conflicts

__device__ __forceinline__ float sgnf(float v) {
    return (v > 0.f) ? 1.f : ((v < 0.f) ? -1.f : 0.f);
}

__global__ __launch_bounds__(32)
void ar_bisect_invert_kernel(const float* __restrict__ y,
                             const float* __restrict__ a,
                             const float* __restrict__ W,
                             float* __restrict__ out) {
    __shared__ float Wl[D64 * WPAD];    // full 64x64 W, row-major padded (16KB)
    __shared__ float Tl[ROWS * WPAD];   // tanh(x) per (row, coord)
    __shared__ float Yl[ROWS * WPAD];   // y rows for this block
    __shared__ float Xl[ROWS * D64];    // solved roots (staged for coalesced store)
    __shared__ float Cl[ROWS * 16];     // per-tile coupling term c (from WMMA)
    __shared__ float SPl[D64];          // softplus(a)

    const int lane = threadIdx.x;          // 0..31
    const int r0   = blockIdx.x * ROWS;    // first batch row of this block

    // ---- stage inputs into LDS (coalesced) ----
    for (int k = lane; k < D64 * D64; k += 32) {
        int r = k >> 6, c = k & 63;
        Wl[r * WPAD + c] = W[k];
    }
    for (int k = lane; k < ROWS * D64; k += 32) {
        int r = k >> 6, c = k & 63;
        Yl[r * WPAD + c] = y[(r0 + r) * D64 + c];
        Tl[r * WPAD + c] = 0.f;            // tanh(0) = 0 (x init = midpoint = 0)
    }
    for (int i = lane; i < D64; i += 32) {
        SPl[i] = log1pf(expf(a[i]));       // softplus, a in [0.5,1.5] -> stable
    }
    __syncthreads();

    const int nl    = lane & 15;           // N index (coord-in-tile / M-row idx)
    const int khalf = (lane >> 4) * 2;     // K base: lanes 0-15 -> {0,1}, 16-31 -> {2,3}
    const int mo    = (lane >> 4) * 8;     // C/D fragment M offset

    // ---- 4 coordinate tiles of 16, blocked triangular sweep ----
    for (int t = 0; t < 4; ++t) {
        // Off-diagonal panels: C[32 rows x 16 coords] += tanh(X_prev) @ W_panel^T
        v8f acc0 = {};                     // M-tile 0 (rows 0..15)
        v8f acc1 = {};                     // M-tile 1 (rows 16..31)
        for (int p = 0; p < t; ++p) {
            #pragma unroll
            for (int q = 0; q < 4; ++q) {  // K = 16 per panel -> 4 chained x4 WMMAs
                const int colb = p * 16 + q * 4;
                // B fragment (4x16): B[k][n] = W[16t+n][colb+k]
                v2f bf;
                bf.x = Wl[(t * 16 + nl) * WPAD + colb + khalf + 0];
                bf.y = Wl[(t * 16 + nl) * WPAD + colb + khalf + 1];
                // A fragments (16x4): A[m][k] = tanh(X)[m][colb+k]
                v2f a0, a1;
                a0.x = Tl[(nl) * WPAD + colb + khalf + 0];
                a0.y = Tl[(nl) * WPAD + colb + khalf + 1];
                a1.x = Tl[(16 + nl) * WPAD + colb + khalf + 0];
                a1.y = Tl[(16 + nl) * WPAD + colb + khalf + 1];
                acc0 = __builtin_amdgcn_wmma_f32_16x16x4_f32(
                           false, a0, false, bf, (short)0, acc0, false, false);
                acc1 = __builtin_amdgcn_wmma_f32_16x16x4_f32(
                           false, a1, false, bf, (short)0, acc1, false, false);
            }
        }
        // Scatter C/D fragments to LDS: lane holds (M = v + 8*(lane>=16), N = lane&15)
        #pragma unroll
        for (int v = 0; v < 8; ++v) {
            Cl[(mo + v) * 16 + nl]        = acc0[v];
            Cl[(16 + mo + v) * 16 + nl]   = acc1[v];
        }
        __syncthreads();

        // Diagonal block: serial over 16 coords, lane = batch row (all 32 busy)
        for (int n = 0; n < 16; ++n) {
            const int i = t * 16 + n;
            float c = Cl[lane * 16 + n];
            for (int j = 0; j < n; ++j)    // strictly-lower part inside the tile
                c += Wl[i * WPAD + t * 16 + j] * Tl[lane * WPAD + t * 16 + j];

            const float yi = Yl[lane * WPAD + i];
            const float sp = SPl[i];       // g(x) = sp*x + c - yi, sp > 0

            // --- interval adaptation (faithful: signs evaluated at PRE-update bounds) ---
            float lo = -10.f, up = 10.f, e = 20.f;
            float sl = sgnf(sp * lo + c - yi);
            float su = sgnf(sp * up + c - yi);
            int it = 0;
            while (sl == su && it < 200) {
                const float nlo = (sl == 1.f) ? (lo - e) : up;
                const float nup = (sl == 1.f) ? lo : (up + e);
                const float s_lo = sgnf(sp * lo + c - yi);  // old bounds, as in reference
                const float s_up = sgnf(sp * up + c - yi);
                lo = nlo; up = nup; e *= 2.f; sl = s_lo; su = s_up; ++it;
            }
            // --- bisection ---
            it = 0;
            while ((up - lo) > 2.0e-6f && it < 200) {
                const float mid = 0.5f * (lo + up);
                const bool pos = (sp * mid + c - yi) >= 0.f;
                lo = pos ? lo : mid;
                up = pos ? mid : up;
                ++it;
            }
            const float root = 0.5f * (lo + up);
            Xl[lane * D64 + i]  = root;
            Tl[lane * WPAD + i] = tanhf(root);
        }
        __syncthreads();
    }

    // ---- coalesced store: Xl row-major == global layout for this block ----
    for (int k = lane; k < ROWS * D64; k += 32)
        out[r0 * D64 + k] = Xl[k];
}

extern "C" void kernel_launch(void* const* d_in, const int* in_sizes, int n_in,
                              void* d_out, int out_size, void* d_ws, size_t ws_size,
                              hipStream_t stream) {
    const float* y = (const float*)d_in[0];   // (512, 64)
    const float* a = (const float*)d_in[1];   // (64,)
    const float* W = (const float*)d_in[2];   // (64, 64), strictly lower-tri
    float* out = (float*)d_out;               // (512, 64)

    const int B = 512;
    ar_bisect_invert_kernel<<<dim3(B / ROWS), dim3(32), 0, stream>>>(y, a, W, out);
}